// HybridVQCLayer_82497731821694
// MI455X (gfx1250) — compile-verified
//
#include <hip/hip_runtime.h>

typedef __attribute__((ext_vector_type(2))) float v2f;
typedef __attribute__((ext_vector_type(8))) float v8f;

#define NQ   10
#define NL   2
#define DIN  256
#define BATCH 4096
#define NPAD 16   // N padded to one WMMA tile

// ======================================================================
// Kernel 0: pad + transpose W[DIN][NQ] -> Wp[NPAD][DIN] (cols >= NQ are 0)
// so the GEMM inner loop needs zero predication and B fragments are
// contiguous 8-byte loads.
// ======================================================================
__global__ __launch_bounds__(256) void vqc_pad_w(
    const float* __restrict__ W, float* __restrict__ Wp)
{
  const int i = blockIdx.x * blockDim.x + threadIdx.x;  // 0 .. NPAD*DIN-1
  const int n = i >> 8;        // 0..15
  const int k = i & (DIN - 1); // 0..255
  Wp[i] = (n < NQ) ? W[k * NQ + n] : 0.0f;
}

// ======================================================================
// Kernel 1: encoder GEMM via fp32 WMMA.
// angles[B, NQ] = x[B, DIN] @ W[DIN, NQ] + b[NQ]
// One wave per 16-row tile; K consumed 4 per WMMA; two accumulators to
// halve the WMMA->WMMA serial chain.
// A 16x4 layout: lane 0-15 holds M=lane, K = 2*half + {0,1} in v0/v1.
// B 4x16 layout: lane holds N = lane&15, K = 2*half + {0,1} in v0/v1.
// C/D 16x16: VGPR v, lanes 0-15 -> M=v, lanes 16-31 -> M=v+8, N=lane&15.
// ======================================================================
__global__ __launch_bounds__(128) void vqc_encoder_wmma(
    const float* __restrict__ x, const float* __restrict__ Wp,
    const float* __restrict__ bias, float* __restrict__ angles)
{
  const int wave = blockIdx.x * (blockDim.x >> 5) + (threadIdx.x >> 5);
  const int lane = threadIdx.x & 31;
  const int half = lane >> 4;
  const int m    = lane & 15;
  const int row0 = wave * 16;

  const float* xr = x + (row0 + m) * DIN + 2 * half;  // A row, this lane's K pair
  const float* wr = Wp + m * DIN + 2 * half;          // B col (transposed), K pair

  v8f acc0 = {}, acc1 = {};
  #pragma unroll 4
  for (int k0 = 0; k0 < DIN; k0 += 8) {
    v2f a0 = *(const v2f*)(xr + k0);
    v2f b0 = *(const v2f*)(wr + k0);
    v2f a1 = *(const v2f*)(xr + k0 + 4);
    v2f b1 = *(const v2f*)(wr + k0 + 4);
    acc0 = __builtin_amdgcn_wmma_f32_16x16x4_f32(false, a0, false, b0,
                                                 (short)0, acc0, false, false);
    acc1 = __builtin_amdgcn_wmma_f32_16x16x4_f32(false, a1, false, b1,
                                                 (short)0, acc1, false, false);
  }

  if (m < NQ) {
    const float bv = bias[m];
    #pragma unroll
    for (int v = 0; v < 8; ++v) {
      const int row = row0 + v + 8 * half;
      angles[row * NQ + m] = acc0[v] + acc1[v] + bv;
    }
  }
}

// ======================================================================
// Kernel 2: 10-qubit statevector simulation, one wave32 per sample.
// 1024 amplitudes: bits 0-4 of index = lane-local slot (re[32]/im[32]
// in VGPRs), bits 5-9 = lane id. Cross-lane gates use __shfl_xor.
// ======================================================================

template<int Q>
__device__ __forceinline__ void ry_local(float re[32], float im[32], float c, float s) {
  #pragma unroll
  for (int j0 = 0; j0 < 32; ++j0) {
    if (j0 & (1 << Q)) continue;
    const int j1 = j0 | (1 << Q);
    const float r0 = re[j0], i0 = im[j0], r1 = re[j1], i1 = im[j1];
    re[j0] = c * r0 - s * r1;
    im[j0] = c * i0 - s * i1;
    re[j1] = s * r0 + c * r1;
    im[j1] = s * i0 + c * i1;
  }
}

template<int Q>
__device__ __forceinline__ void ry_cross(float re[32], float im[32], float c, float s, int lane) {
  const int   lm = 1 << (Q - 5);
  const float sp = (lane & lm) ? s : -s;   // coefficient applied to partner amp
  #pragma unroll
  for (int j = 0; j < 32; ++j) {
    const float pr = __shfl_xor(re[j], lm, 32);
    const float pi = __shfl_xor(im[j], lm, 32);
    re[j] = c * re[j] + sp * pr;
    im[j] = c * im[j] + sp * pi;
  }
}

template<int Q>
__device__ __forceinline__ void rz_local(float re[32], float im[32], float cr, float ci) {
  #pragma unroll
  for (int j = 0; j < 32; ++j) {
    const float ph = ((j >> Q) & 1) ? ci : -ci;
    const float r = re[j], i = im[j];
    re[j] = r * cr - i * ph;
    im[j] = r * ph + i * cr;
  }
}

template<int Q>
__device__ __forceinline__ void rz_cross(float re[32], float im[32], float cr, float ci, int lane) {
  const float ph = ((lane >> (Q - 5)) & 1) ? ci : -ci;
  #pragma unroll
  for (int j = 0; j < 32; ++j) {
    const float r = re[j], i = im[j];
    re[j] = r * cr - i * ph;
    im[j] = r * ph + i * cr;
  }
}

template<int C, int T>
__device__ __forceinline__ void cnot_ll(float re[32], float im[32]) {
  #pragma unroll
  for (int j = 0; j < 32; ++j) {
    if (((j >> C) & 1) && !((j >> T) & 1)) {
      const int k = j | (1 << T);
      const float tr = re[j]; re[j] = re[k]; re[k] = tr;
      const float ti = im[j]; im[j] = im[k]; im[k] = ti;
    }
  }
}

// control = qubit 4 (local MSB), target = qubit 5 (lane bit 0):
// for local slots with bit4 set, swap amplitudes with lane^1.
__device__ __forceinline__ void cnot_45(float re[32], float im[32]) {
  #pragma unroll
  for (int j = 16; j < 32; ++j) {
    re[j] = __shfl_xor(re[j], 1, 32);
    im[j] = __shfl_xor(im[j], 1, 32);
  }
}

template<int C, int T>
__device__ __forceinline__ void cnot_cc(float re[32], float im[32], int lane) {
  const int  tm = 1 << (T - 5);
  const bool cb = (lane >> (C - 5)) & 1;
  #pragma unroll
  for (int j = 0; j < 32; ++j) {
    const float pr = __shfl_xor(re[j], tm, 32);
    const float pi = __shfl_xor(im[j], tm, 32);
    re[j] = cb ? pr : re[j];
    im[j] = cb ? pi : im[j];
  }
}

__global__ __launch_bounds__(256) void vqc_circuit(
    const float* __restrict__ angles, const float* __restrict__ theta,
    float* __restrict__ out)
{
  const int wave = blockIdx.x * (blockDim.x >> 5) + (threadIdx.x >> 5);
  const int lane = threadIdx.x & 31;
  const float* ang = angles + wave * NQ;

  float re[32], im[32];
  #pragma unroll
  for (int j = 0; j < 32; ++j) { re[j] = 0.0f; im[j] = 0.0f; }
  if (lane == 0) re[0] = 1.0f;

  #define G_RY_L(Q, A) { const float _h = (A) * 0.5f; ry_local<Q>(re, im, __cosf(_h), __sinf(_h)); }
  #define G_RY_C(Q, A) { const float _h = (A) * 0.5f; ry_cross<Q>(re, im, __cosf(_h), __sinf(_h), lane); }
  #define G_RZ_L(Q, A) { const float _h = (A) * 0.5f; rz_local<Q>(re, im, __cosf(_h), __sinf(_h)); }
  #define G_RZ_C(Q, A) { const float _h = (A) * 0.5f; rz_cross<Q>(re, im, __cosf(_h), __sinf(_h), lane); }

  // ---- data encoding: per-sample RY on each qubit ----
  G_RY_L(0, ang[0]); G_RY_L(1, ang[1]); G_RY_L(2, ang[2]);
  G_RY_L(3, ang[3]); G_RY_L(4, ang[4]);
  G_RY_C(5, ang[5]); G_RY_C(6, ang[6]); G_RY_C(7, ang[7]);
  G_RY_C(8, ang[8]); G_RY_C(9, ang[9]);

  // ---- variational layers ----
  for (int l = 0; l < NL; ++l) {
    const float* th = theta + l * NQ * 2;
    G_RY_L(0, th[0]);  G_RZ_L(0, th[1]);
    G_RY_L(1, th[2]);  G_RZ_L(1, th[3]);
    G_RY_L(2, th[4]);  G_RZ_L(2, th[5]);
    G_RY_L(3, th[6]);  G_RZ_L(3, th[7]);
    G_RY_L(4, th[8]);  G_RZ_L(4, th[9]);
    G_RY_C(5, th[10]); G_RZ_C(5, th[11]);
    G_RY_C(6, th[12]); G_RZ_C(6, th[13]);
    G_RY_C(7, th[14]); G_RZ_C(7, th[15]);
    G_RY_C(8, th[16]); G_RZ_C(8, th[17]);
    G_RY_C(9, th[18]); G_RZ_C(9, th[19]);

    // linear entanglers (0,1)...(8,9)
    cnot_ll<0,1>(re, im);
    cnot_ll<1,2>(re, im);
    cnot_ll<2,3>(re, im);
    cnot_ll<3,4>(re, im);
    cnot_45(re, im);
    cnot_cc<5,6>(re, im, lane);
    cnot_cc<6,7>(re, im, lane);
    cnot_cc<7,8>(re, im, lane);
    cnot_cc<8,9>(re, im, lane);
  }

  // ---- measurement: mean_q <Z_q> = sum_i p_i * (NQ - 2*popcount(i)) / NQ ----
  const float plf = 2.0f * (float)__popc(lane);
  float acc = 0.0f;
  #pragma unroll
  for (int j = 0; j < 32; ++j) {
    const float p = re[j] * re[j] + im[j] * im[j];
    acc += p * ((float)(NQ - 2 * __builtin_popcount(j)) - plf);
  }
  acc *= (1.0f / (float)NQ);

  #pragma unroll
  for (int off = 16; off > 0; off >>= 1) acc += __shfl_xor(acc, off, 32);

  // MIN_OUT=-1, MAX_OUT=1 => output == expectation value
  if (lane == 0) out[wave] = acc;
}

// ======================================================================
extern "C" void kernel_launch(void* const* d_in, const int* in_sizes, int n_in,
                              void* d_out, int out_size, void* d_ws, size_t ws_size,
                              hipStream_t stream) {
  const float* x     = (const float*)d_in[0];   // [4096, 256]
  const float* W     = (const float*)d_in[1];   // [256, 10]
  const float* b     = (const float*)d_in[2];   // [10]
  const float* theta = (const float*)d_in[3];   // [2, 10, 2]
  float* out    = (float*)d_out;                // [4096]
  float* angles = (float*)d_ws;                          // [4096, 10]
  float* Wp     = angles + BATCH * NQ;                   // [16, 256] padded W^T

  // Pad/transpose W (one-shot, 4096 elements)
  vqc_pad_w<<<(NPAD * DIN) / 256, 256, 0, stream>>>(W, Wp);
  // GEMM: 256 row-tiles of 16, 4 waves (128 threads) per block -> 64 blocks
  vqc_encoder_wmma<<<(BATCH / 16) / 4, 128, 0, stream>>>(x, Wp, b, angles);
  // Circuit: one wave per sample, 8 waves (256 threads) per block
  vqc_circuit<<<BATCH / 8, 256, 0, stream>>>(angles, theta, out);
}